// SGAT32_3496103379555
// MI455X (gfx1250) — compile-verified
//
#include <hip/hip_runtime.h>
#include <hip/hip_bf16.h>

// SuperGAT (MX) stack for gfx1250 (MI455X, wave32).
// - Dense node transforms: V_WMMA_F32_16X16X4_F32 (fp32-exact matrix path),
//   B operand staged in LDS (one copy per block).
// - Edge stage: 2 passes, 8 lanes per edge so every feature-row gather is one
//   contiguous 128B (cacheline) access; dot reduced with __shfl_xor.
// - Softmax denominator folded into aggregation; normalize+bias+ReLU per node.

#define C32 32

typedef float v2f __attribute__((ext_vector_type(2)));
typedef float v8f __attribute__((ext_vector_type(8)));

// ---------------------------------------------------------------------------
// GEMM: out[M x 32] = A[M x K] @ W[K x 32] (+ bias[32] if use_bias)
// One wave computes a 16-row x 32-col tile as two 16x16 f32 WMMA accumulators.
// A-matrix VGPR layout (ISA 7.12.2, 32-bit A 16x4): lanes 0-15 carry K=0,1 in
// v[0],v[1]; lanes 16-31 carry K=2,3. B/C/D are row-striped across lanes.
// W (<=128x32 = 16KB) is staged into LDS once per block.
// ---------------------------------------------------------------------------
__global__ void gemm_wmma_f32(const float* __restrict__ A,
                              const float* __restrict__ W,
                              const float* __restrict__ bias,
                              float* __restrict__ out,
                              int M, int K, int use_bias) {
  __shared__ float Wl[128 * C32];            // max K = 128
  for (int i = threadIdx.x; i < K * C32; i += blockDim.x) Wl[i] = W[i];
  __syncthreads();                           // all waves reach this barrier

  const int lane = threadIdx.x & 31;
  const int wave = threadIdx.x >> 5;
  const int wpb  = blockDim.x >> 5;
  const long tile = (long)blockIdx.x * wpb + wave;
  const int m0 = (int)(tile * 16);
  if (m0 >= M) return;                       // wave-uniform exit (EXEC stays full)

  const int half = lane >> 4;                // 0: K pair {0,1}, 1: pair {2,3}
  const int l15  = lane & 15;
  int arow = m0 + l15;
  if (arow >= M) arow = M - 1;               // clamp loads; stores guarded later

  v8f acc0, acc1;
  const float bv0 = use_bias ? bias[l15]      : 0.0f;
  const float bv1 = use_bias ? bias[16 + l15] : 0.0f;
#pragma unroll
  for (int v = 0; v < 8; ++v) { acc0[v] = bv0; acc1[v] = bv1; }

  const float* arow_p = A + (long)arow * K;
  for (int k = 0; k < K; k += 4) {
    const int kk = k + 2 * half;
    v2f a;                                   // A 16x4 fragment (global stream)
    a.x = arow_p[kk];
    a.y = arow_p[kk + 1];
    v2f b0, b1;                              // B 4x16 fragments (from LDS)
    b0.x = Wl[kk * C32 + l15];
    b0.y = Wl[(kk + 1) * C32 + l15];
    b1.x = Wl[kk * C32 + 16 + l15];
    b1.y = Wl[(kk + 1) * C32 + 16 + l15];
    acc0 = __builtin_amdgcn_wmma_f32_16x16x4_f32(false, a, false, b0,
                                                 (short)0, acc0, false, false);
    acc1 = __builtin_amdgcn_wmma_f32_16x16x4_f32(false, a, false, b1,
                                                 (short)0, acc1, false, false);
  }

  // D layout: VGPR v, lanes 0-15 -> row m0+v, lanes 16-31 -> row m0+v+8
#pragma unroll
  for (int v = 0; v < 8; ++v) {
    const int row = m0 + v + 8 * half;
    if (row < M) {
      out[(long)row * C32 + l15]      = acc0[v];
      out[(long)row * C32 + 16 + l15] = acc1[v];
    }
  }
}

// ---------------------------------------------------------------------------
// Per-node prep: aL[n] = hlin[n].att_l, aR[n] = hlin[n].att_r, and reset the
// per-layer accumulators (amax encoded-uint, denom, acc vector).
// ---------------------------------------------------------------------------
__global__ void node_prep(const float* __restrict__ hl,
                          const float* __restrict__ al,
                          const float* __restrict__ ar,
                          float* __restrict__ aL, float* __restrict__ aR,
                          unsigned* __restrict__ amax,
                          float* __restrict__ denom,
                          float* __restrict__ acc, int N) {
  const int n = blockIdx.x * blockDim.x + threadIdx.x;
  if (n >= N) return;
  const float4* h4 = (const float4*)(hl + (long)n * C32);
  const float4* a4 = (const float4*)al;
  const float4* r4 = (const float4*)ar;
  float4* acc4 = (float4*)(acc + (long)n * C32);
  float sl = 0.0f, sr = 0.0f;
#pragma unroll
  for (int i = 0; i < 8; ++i) {
    float4 h = h4[i], a = a4[i], r = r4[i];
    sl += h.x * a.x + h.y * a.y + h.z * a.z + h.w * a.w;
    sr += h.x * r.x + h.y * r.y + h.z * r.z + h.w * r.w;
    acc4[i] = make_float4(0.f, 0.f, 0.f, 0.f);
  }
  aL[n] = sl;
  aR[n] = sr;
  amax[n] = 0u;        // minimum of the monotone float encoding
  denom[n] = 0.0f;
}

// Monotone float <-> uint mapping for atomic max over signed floats.
__device__ __forceinline__ unsigned fenc(float f) {
  int i = __float_as_int(f);
  unsigned u = (unsigned)i;
  return (i >= 0) ? (u | 0x80000000u) : ~u;
}
__device__ __forceinline__ float fdec(unsigned u) {
  return (u & 0x80000000u) ? __int_as_float((int)(u & 0x7FFFFFFFu))
                           : __int_as_float((int)~u);
}

// ---------------------------------------------------------------------------
// Edge pass 1 (8 lanes per edge): alpha_e = LeakyReLU((aL[s]+aR[d]) *
// sigmoid(<hi,hj>)); store alpha, atomic-max per destination.
// Each lane loads one contiguous float4 of the 128B feature row.
// ---------------------------------------------------------------------------
__global__ void edge_max(const int* __restrict__ ei,
                         const float* __restrict__ hl,
                         const float* __restrict__ aL,
                         const float* __restrict__ aR,
                         float* __restrict__ alpha,
                         unsigned* __restrict__ amax,
                         int E, int Etot) {
  const long t = (long)blockIdx.x * blockDim.x + threadIdx.x;
  const int  sub = (int)(t & 7);
  const long e   = t >> 3;
  if (e >= Etot) return;                 // whole 8-lane group exits together
  __builtin_prefetch(ei + e + 8192, 0, 0);   // speculative stream prefetch
  int s, d;
  if (e < E) { s = ei[e]; d = ei[E + e]; } else { s = d = (int)(e - E); }
  const float4 a = ((const float4*)(hl + (long)s * C32))[sub];
  const float4 b = ((const float4*)(hl + (long)d * C32))[sub];
  float dot = a.x * b.x + a.y * b.y + a.z * b.z + a.w * b.w;
  dot += __shfl_xor(dot, 1, 32);         // reduce across the 8-lane group
  dot += __shfl_xor(dot, 2, 32);
  dot += __shfl_xor(dot, 4, 32);
  const float gate = 1.0f / (1.0f + __expf(-dot));     // sigmoid(<hi,hj>)
  float al = (aL[s] + aR[d]) * gate;
  al = (al > 0.0f) ? al : 0.2f * al;                   // LeakyReLU(0.2)
  if (sub == 0) {
    alpha[e] = al;
    atomicMax(&amax[d], fenc(al));
  }
}

// ---------------------------------------------------------------------------
// Edge pass 2 (8 lanes per edge): ea = exp(alpha - max[dst]);
// denom[dst] += ea (one lane); acc[dst] += ea * hlin[src] (lane-coalesced
// f32 atomics into a single 128B line per edge).
// ---------------------------------------------------------------------------
__global__ void edge_sum(const int* __restrict__ ei,
                         const float* __restrict__ hl,
                         const float* __restrict__ alpha,
                         const unsigned* __restrict__ amax,
                         float* __restrict__ denom,
                         float* __restrict__ acc,
                         int E, int Etot) {
  const long t = (long)blockIdx.x * blockDim.x + threadIdx.x;
  const int  sub = (int)(t & 7);
  const long e   = t >> 3;
  if (e >= Etot) return;
  __builtin_prefetch(ei + e + 8192, 0, 0);
  int s, d;
  if (e < E) { s = ei[e]; d = ei[E + e]; } else { s = d = (int)(e - E); }
  const float ea = __expf(alpha[e] - fdec(amax[d]));
  if (sub == 0) atomicAdd(&denom[d], ea);
  const float4 h = ((const float4*)(hl + (long)s * C32))[sub];
  float* ap = acc + (long)d * C32 + sub * 4;
  atomicAdd(&ap[0], ea * h.x);
  atomicAdd(&ap[1], ea * h.y);
  atomicAdd(&ap[2], ea * h.z);
  atomicAdd(&ap[3], ea * h.w);
}

// ---------------------------------------------------------------------------
// Node finish: h_next = ReLU(acc/(denom+1e-16) + b). One thread per (node,ch).
// ---------------------------------------------------------------------------
__global__ void node_finish(const float* __restrict__ acc,
                            const float* __restrict__ denom,
                            const float* __restrict__ b,
                            float* __restrict__ hout, int N) {
  const long t = (long)blockIdx.x * blockDim.x + threadIdx.x;
  const long n = t >> 5;
  const int  c = (int)(t & 31);
  if (n >= N) return;
  float v = acc[n * C32 + c] / (denom[n] + 1e-16f) + b[c];
  hout[n * C32 + c] = (v > 0.0f) ? v : 0.0f;
}

// ---------------------------------------------------------------------------
extern "C" void kernel_launch(void* const* d_in, const int* in_sizes, int n_in,
                              void* d_out, int out_size, void* d_ws, size_t ws_size,
                              hipStream_t stream) {
  const float* x    = (const float*)d_in[0];
  const int*   ei   = (const int*)  d_in[1];
  const float* W0   = (const float*)d_in[2];
  const float* b0   = (const float*)d_in[3];
  const float* Ws   = (const float*)d_in[4];
  const float* bs   = (const float*)d_in[5];
  const float* attl = (const float*)d_in[6];
  const float* attr = (const float*)d_in[7];
  const float* Wout = (const float*)d_in[8];
  const float* bout = (const float*)d_in[9];

  const int IN   = 128;
  const int N    = in_sizes[0] / IN;
  const int E    = in_sizes[1] / 2;
  const int L    = in_sizes[4] / (C32 * C32);
  const int Etot = E + N;

  // Workspace layout (all f32-sized slots; ~53 MB total)
  float* bufH  = (float*)d_ws;                 // [N,32] current features
  float* bufL  = bufH + (size_t)N * C32;       // [N,32] h @ W (per layer)
  float* acc   = bufL + (size_t)N * C32;       // [N,32] numerator accumulator
  float* aL    = acc  + (size_t)N * C32;       // [N]
  float* aR    = aL + N;                       // [N]
  float* denom = aR + N;                       // [N]
  unsigned* amax = (unsigned*)(denom + N);     // [N]
  float* alpha = (float*)(amax + N);           // [Etot]

  const dim3 blk(256);
  const int gemmBlocks = (((N + 15) / 16) + 7) / 8;          // 8 waves/block
  const int nodeBlocks = (N + 255) / 256;
  const int edgeBlocks = (int)(((long)Etot * 8 + 255) / 256);
  const int ncBlocks   = (int)(((long)N * C32 + 255) / 256);

  // h0 = x @ W0 + b0
  gemm_wmma_f32<<<gemmBlocks, blk, 0, stream>>>(x, W0, b0, bufH, N, IN, 1);

  for (int l = 0; l < L; ++l) {
    const float* Wl = Ws + (size_t)l * C32 * C32;
    const float* bl = bs + (size_t)l * C32;
    const float* al = attl + (size_t)l * C32;
    const float* ar = attr + (size_t)l * C32;
    // hlin = h @ W (bias is applied post-aggregation, per reference)
    gemm_wmma_f32<<<gemmBlocks, blk, 0, stream>>>(bufH, Wl, bl, bufL, N, C32, 0);
    node_prep<<<nodeBlocks, blk, 0, stream>>>(bufL, al, ar, aL, aR, amax, denom, acc, N);
    edge_max<<<edgeBlocks, blk, 0, stream>>>(ei, bufL, aL, aR, alpha, amax, E, Etot);
    edge_sum<<<edgeBlocks, blk, 0, stream>>>(ei, bufL, alpha, amax, denom, acc, E, Etot);
    node_finish<<<ncBlocks, blk, 0, stream>>>(acc, denom, bl, bufH, N);
  }

  // out = h @ Wout + bout
  gemm_wmma_f32<<<gemmBlocks, blk, 0, stream>>>(bufH, Wout, bout, (float*)d_out, N, C32, 1);
}